// GeomGCNEncoder_61220463837362
// MI455X (gfx1250) — compile-verified
//
#include <hip/hip_runtime.h>
#include <stdint.h>

typedef __attribute__((ext_vector_type(2))) float v2f;
typedef __attribute__((ext_vector_type(8))) float v8f;

#define DIM 128
#define TILE_M 16

// ---------------- prep kernels ----------------

__global__ void k_zero_f32(float* __restrict__ p, int n) {
  int i = blockIdx.x * blockDim.x + threadIdx.x;
  if (i < n) p[i] = 0.0f;
}

__global__ void k_degree(const long long* __restrict__ src, int E, float* __restrict__ deg) {
  int e = blockIdx.x * blockDim.x + threadIdx.x;
  if (e < E) atomicAdd(&deg[(int)src[e]], 1.0f);  // integer counts: exact & deterministic
}

__global__ void k_dinv(float* __restrict__ deg, int n) {
  int i = blockIdx.x * blockDim.x + threadIdx.x;
  if (i < n) deg[i] = rsqrtf(deg[i]);             // in place: deg -> d^{-1/2}
}

// src is sorted ascending: rowptr[n] = lower_bound(src, n)
__global__ void k_rowptr(const long long* __restrict__ src, int E, int N, int* __restrict__ rowptr) {
  int n = blockIdx.x * blockDim.x + threadIdx.x;
  if (n <= N) {
    int lo = 0, hi = E;
    while (lo < hi) { int mid = (lo + hi) >> 1; if ((int)src[mid] < n) lo = mid + 1; else hi = mid; }
    rowptr[n] = lo;
  }
}

// pack (dst, weight) into one 8-byte record per edge: single b64 load in the hot loop
__global__ void k_edgeprep(const long long* __restrict__ src, const long long* __restrict__ dst,
                           const float* __restrict__ dinv, int E, int2* __restrict__ epack) {
  int e = blockIdx.x * blockDim.x + threadIdx.x;
  if (e < E) {
    int s = (int)src[e], d = (int)dst[e];
    float w = dinv[s] * dinv[d];
    epack[e] = make_int2(d, __float_as_int(w));
  }
}

// 128x128 transpose: Wt[c][k] = W[k][c], so B fragments (W[k][c], W[k+1][c]) are contiguous b64
__global__ __launch_bounds__(256)
void k_transpose128(const float* __restrict__ W, float* __restrict__ Wt) {
  __shared__ float t[16][17];
  const int bx = blockIdx.x & 7, by = blockIdx.x >> 3;     // 8x8 tiles of 16x16
  const int tx = threadIdx.x & 15, ty = threadIdx.x >> 4;
  t[ty][tx] = W[(size_t)(by * 16 + ty) * DIM + bx * 16 + tx];
  __syncthreads();
  Wt[(size_t)(bx * 16 + ty) * DIM + by * 16 + tx] = t[tx][ty];
}

// ---------------- fused SpMM + GEMM layer ----------------
// Block: 256 threads = 8 waves; 16 output rows per block.
// Phase 1: agg tile (16 x 128) in LDS. Phase 2: tile @ W^T-staged weights via WMMA f32.

__global__ __launch_bounds__(256)
void k_layer(const float* __restrict__ hin,      // N x 128
             const int2*  __restrict__ epack,    // E  (dst, weight-bits)
             const int*   __restrict__ rowptr,   // N+1
             const float* __restrict__ Wt,       // 128 x 128, Wt[c][k] = W[k][c]
             const float* __restrict__ bias,     // 128
             float*       __restrict__ hout,     // N x 128
             int N, int do_relu)
{
  __shared__ float sA[TILE_M][DIM + 4];          // row stride 132 floats: 16B aligned, bank-safe
  const int tid  = threadIdx.x;
  const int lane = tid & 31;
  const int wave = tid >> 5;
  const int row0 = blockIdx.x * TILE_M;

  // Hint this wave's weight strip into the caches (global_prefetch_b8).
  __builtin_prefetch(Wt + (size_t)(wave * 16) * DIM + lane * 4, 0, 3);

  // ---- Phase 1: SpMM. 8 waves x 2 rows; lane owns 4 contiguous channels -> one b128 per edge.
  for (int rr = 0; rr < 2; ++rr) {
    const int lr = wave * 2 + rr;
    const int r  = row0 + lr;
    float a0 = 0.f, a1 = 0.f, a2 = 0.f, a3 = 0.f;
    if (r < N) {
      const int e0 = __builtin_amdgcn_readfirstlane(rowptr[r]);
      const int e1 = __builtin_amdgcn_readfirstlane(rowptr[r + 1]);
      for (int e = e0; e < e1; ++e) {
        const int2  ed = epack[e];
        const int   d  = __builtin_amdgcn_readfirstlane(ed.x);                  // SGPR base
        const float w  = __int_as_float(__builtin_amdgcn_readfirstlane(ed.y));  // scalar FMA operand
        const float4 hv = *(const float4*)(hin + (size_t)d * DIM + lane * 4);   // global_load_b128
        a0 = fmaf(w, hv.x, a0);
        a1 = fmaf(w, hv.y, a1);
        a2 = fmaf(w, hv.z, a2);
        a3 = fmaf(w, hv.w, a3);
      }
    }
    *(float4*)&sA[lr][lane * 4] = make_float4(a0, a1, a2, a3);
  }
  __syncthreads();

  // ---- Phase 2: 16x16 tile per wave via V_WMMA_F32_16X16X4_F32, K-loop over 128.
  // A (16x4 f32): lanes 0-15 = M rows; VGPR0/1 = K{0,1} (lo half) / K{2,3} (hi half).
  // B (4x16 f32): lanes hold N cols; same K split across lane halves.  Wt makes B a b64 load.
  const int col0 = wave * 16;
  const int m16  = lane & 15;
  const int kh   = (lane >> 4) << 1;             // 0 for lanes 0-15, 2 for lanes 16-31
  const float* __restrict__ wrow = Wt + (size_t)(col0 + m16) * DIM + kh;
  v8f c = {0.f, 0.f, 0.f, 0.f, 0.f, 0.f, 0.f, 0.f};
  #pragma unroll 4
  for (int k = 0; k < DIM; k += 4) {
    v2f a, b;
    a[0] = sA[m16][k + kh];
    a[1] = sA[m16][k + kh + 1];
    b = *(const v2f*)(wrow + k);                 // contiguous, 8B aligned (kh even)
    c = __builtin_amdgcn_wmma_f32_16x16x4_f32(false, a, false, b, (short)0, c, false, false);
  }

  // C/D layout: VGPR v holds M = v + (lane>=16 ? 8 : 0), N = lane & 15.
  const float bcol  = bias[col0 + m16];
  const int   mbase = (lane >> 4) * 8;
  float* __restrict__ op = hout + (size_t)(row0 + mbase) * DIM + col0 + m16;
  if (row0 + TILE_M <= N) {                      // uniform branch: full tile (always, N%16==0)
    if (do_relu) {
      #pragma unroll
      for (int v = 0; v < 8; ++v) op[(size_t)v * DIM] = fmaxf(c[v] + bcol, 0.0f);
    } else {
      #pragma unroll
      for (int v = 0; v < 8; ++v) op[(size_t)v * DIM] = c[v] + bcol;
    }
  } else {                                       // ragged tail fallback
    #pragma unroll
    for (int v = 0; v < 8; ++v) {
      const int r = row0 + mbase + v;
      float val = c[v] + bcol;
      if (do_relu) val = fmaxf(val, 0.0f);
      if (r < N) op[(size_t)v * DIM] = val;
    }
  }
}

// ---------------- host launch ----------------

extern "C" void kernel_launch(void* const* d_in, const int* in_sizes, int n_in,
                              void* d_out, int out_size, void* d_ws, size_t ws_size,
                              hipStream_t stream) {
  const float*     x   = (const float*)d_in[0];
  const long long* ei  = (const long long*)d_in[1];
  const float*     W1  = (const float*)d_in[2];
  const float*     b1  = (const float*)d_in[3];
  const float*     W2  = (const float*)d_in[4];
  const float*     b2  = (const float*)d_in[5];
  float*           out = (float*)d_out;

  const int N = in_sizes[0] / DIM;
  const int E = in_sizes[1] / 2;
  const long long* src = ei;
  const long long* dst = ei + E;

  // carve workspace (256B aligned)
  uintptr_t p = (uintptr_t)d_ws;
  auto carve = [&](size_t bytes) -> void* {
    uintptr_t r = (p + 255) & ~(uintptr_t)255;
    p = r + bytes;
    return (void*)r;
  };
  float* dinv   = (float*)carve(sizeof(float) * (size_t)N);
  int*   rowptr = (int*)  carve(sizeof(int)   * (size_t)(N + 1));
  int2*  epack  = (int2*) carve(sizeof(int2)  * (size_t)E);
  float* W1t    = (float*)carve(sizeof(float) * (size_t)DIM * DIM);
  float* W2t    = (float*)carve(sizeof(float) * (size_t)DIM * DIM);
  float* hbuf   = (float*)carve(sizeof(float) * (size_t)N * DIM);
  (void)ws_size; (void)n_in; (void)out_size;

  const int T = 256;
  k_zero_f32   <<<(N + T - 1) / T, T, 0, stream>>>(dinv, N);
  k_degree     <<<(E + T - 1) / T, T, 0, stream>>>(src, E, dinv);
  k_dinv       <<<(N + T - 1) / T, T, 0, stream>>>(dinv, N);
  k_rowptr     <<<(N + 1 + T - 1) / T, T, 0, stream>>>(src, E, N, rowptr);
  k_edgeprep   <<<(E + T - 1) / T, T, 0, stream>>>(src, dst, dinv, E, epack);
  k_transpose128<<<64, 256, 0, stream>>>(W1, W1t);   // only rows 0..127 of W matter
  k_transpose128<<<64, 256, 0, stream>>>(W2, W2t);

  const int tiles = (N + TILE_M - 1) / TILE_M;
  k_layer<<<tiles, 256, 0, stream>>>(x,    epack, rowptr, W1t, b1, hbuf, N, 1);
  k_layer<<<tiles, 256, 0, stream>>>(hbuf, epack, rowptr, W2t, b2, out,  N, 0);
}